// TokenSenEmbedding_74053826118053
// MI455X (gfx1250) — compile-verified
//
#include <hip/hip_runtime.h>

// Dual-table embedding gather:
//   out[r, :] = (src[r] < VOCAB) ? tok_emb[src[r], :] * sqrt(768)
//                                : sen_emb[src[r]-VOCAB, :]
// r = 0..32767 (flattened [1024, 32]), EMB = 768 floats = 3072 bytes/row.
//
// Pure HBM-bandwidth-bound (~192 MB total traffic -> ~8.3 us at 23.3 TB/s).
// Strategy: one wave32 per row; 6x global_load_b128 + scale + 6x
// non-temporal global_store_b128 per lane-group. No WMMA: there is no
// matrix op in this workload.

typedef float v4f __attribute__((ext_vector_type(4)));

#define VOCAB      100000
#define EMB        768
#define SQRT_EMB   27.712812921102035f   // sqrt(768) = 16*sqrt(3)
#define WAVES_PER_BLOCK 8
#define BLOCK_THREADS   (WAVES_PER_BLOCK * 32)

__global__ __launch_bounds__(BLOCK_THREADS)
void tok_sen_gather_kernel(const int*   __restrict__ src,
                           const float* __restrict__ tok_emb,
                           const float* __restrict__ sen_emb,
                           float*       __restrict__ out,
                           int nrows)
{
    // One wave (32 lanes) handles one output row of 768 floats.
    const int wave = blockIdx.x * WAVES_PER_BLOCK + (threadIdx.x >> 5);
    if (wave >= nrows) return;
    const int lane = threadIdx.x & 31;

    // All 32 lanes read the same id -> single coalesced request.
    const int id = src[wave];

    // Branchless table/scale select (wave-uniform in practice).
    const bool is_tok = (id < VOCAB);
    const int  row    = is_tok ? id : (id - VOCAB);
    const float* base = is_tok ? (tok_emb + (size_t)row * EMB)
                               : (sen_emb + (size_t)row * EMB);
    const float scale = is_tok ? SQRT_EMB : 1.0f;

    const v4f* __restrict__ src_row = (const v4f*)base;           // 192 float4s
    v4f*       __restrict__ dst_row = (v4f*)(out + (size_t)wave * EMB);

    // 6 iterations x 32 lanes x 16B = 3072 B. Issue all loads first so the
    // compiler keeps 6 global_load_b128 in flight per wave, then scale and
    // stream out with non-temporal stores (write-once output; preserve L2
    // for embedding-row reuse).
    v4f v[6];
#pragma unroll
    for (int i = 0; i < 6; ++i)
        v[i] = src_row[i * 32 + lane];

#pragma unroll
    for (int i = 0; i < 6; ++i) {
        v4f t = v[i] * scale;
        __builtin_nontemporal_store(t, dst_row + i * 32 + lane);
    }
}

extern "C" void kernel_launch(void* const* d_in, const int* in_sizes, int n_in,
                              void* d_out, int out_size, void* d_ws, size_t ws_size,
                              hipStream_t stream)
{
    const int*   src     = (const int*)  d_in[0];  // [1024*32] ids
    const float* tok_emb = (const float*)d_in[1];  // [100000, 768]
    const float* sen_emb = (const float*)d_in[2];  // [50000, 768]
    float*       out     = (float*)      d_out;    // [1024*32, 768]

    const int nrows  = in_sizes[0];                        // 32768 rows
    const int blocks = (nrows + WAVES_PER_BLOCK - 1) / WAVES_PER_BLOCK;

    tok_sen_gather_kernel<<<blocks, BLOCK_THREADS, 0, stream>>>(
        src, tok_emb, sen_emb, out, nrows);
}